// NbitTreeDecoder_11029476016272
// MI455X (gfx1250) — compile-verified
//
#include <hip/hip_runtime.h>
#include <hip/hip_bf16.h>

// CDNA5 / gfx1250: wave32, WMMA f32 16x16x4.
typedef __attribute__((ext_vector_type(2))) float v2f;
typedef __attribute__((ext_vector_type(8))) float v8f;

#define N_POINTS (1u << 20)   // 4^10 leaves
#define BLOCK 256u

// Decode the 4-ary occupancy tree, bit-reverse the 30-bit code, and apply
// scale/offset via V_WMMA_F32_16X16X4_F32:  D = [diag(s)|o](16x4) x [xf;1](4x16).
__global__ __launch_bounds__(256) void nbit_tree_decode_wmma(
    const int*   __restrict__ flags,
    const float* __restrict__ offs,
    const float* __restrict__ scal,
    float*       __restrict__ out)
{
    const unsigned tid      = threadIdx.x;
    const unsigned p        = blockIdx.x * BLOCK + tid;   // leaf id
    const int      lane     = (int)(tid & 31u);
    const unsigned waveBase = p & ~31u;                   // first leaf of this wave

    // ---- tree walk: 10 levels, 3 bits per level --------------------------
    unsigned X = 0u;
#pragma unroll
    for (int t = 0; t < 10; ++t) {
        const unsigned pos[10] = {0u, 1u, 5u, 21u, 85u, 341u,
                                  1365u, 5461u, 21845u, 87381u}; // (4^t-1)/3
        unsigned idx = pos[t] + (p >> (unsigned)(2 * (10 - t)));
        if (t <= 7) {
            // node index is identical across the wave for t<=7 -> scalar load
            idx = (unsigned)__builtin_amdgcn_readfirstlane((int)idx);
        }
        unsigned f = (unsigned)flags[idx];
        unsigned d = (p >> (unsigned)(2 * (9 - t))) & 3u; // which child (0..3)
        // position of the d-th set bit: drop d lowest set bits, then ctz
        unsigned m = f;
        m = (d > 0u) ? (m & (m - 1u)) : m;
        m = (d > 1u) ? (m & (m - 1u)) : m;
        m = (d > 2u) ? (m & (m - 1u)) : m;
        X = (X << 3) | (unsigned)__builtin_ctz(m);
    }

    // ---- 30-bit reversal permutation (PERMUTE[b] = 29-b) -----------------
    const unsigned Y = __brev(X) >> 2;

    const float xf0 = (float)(Y & 1023u);
    const float xf1 = (float)((Y >> 10) & 1023u);
    const float xf2 = (float)(Y >> 20);

    // ---- realize: out[p][d] = xf_d * scale[d] + offset[d] via f32 WMMA ---
    const float s0 = scal[0], s1 = scal[1], s2 = scal[2];
    const float o0 = offs[0], o1 = offs[1], o2 = offs[2];

    const bool lo = (lane < 16);
    const int  n  = lane & 15;

    // A (16x4): A[m][0..2] = diag(scale), A[m][3] = offset[m]  (rows >=3 zero)
    // layout: lanes 0-15 hold K=0,1 in {v0,v1}; lanes 16-31 hold K=2,3
    float a0, a1;
    if (lo) { a0 = (n == 0) ? s0 : 0.0f;  a1 = (n == 1) ? s1 : 0.0f; }
    else    { a0 = (n == 2) ? s2 : 0.0f;
              a1 = (n == 0) ? o0 : (n == 1) ? o1 : (n == 2) ? o2 : 0.0f; }
    v2f A; A.x = a0; A.y = a1;

#pragma unroll
    for (int g = 0; g < 2; ++g) {                 // two 16-leaf groups / wave
        // B (4x16): B[k][n] = xf_k[leaf g*16+n] (k<3), B[3][n] = 1.0
        float b0, b1;
        if (g == 0) {
            float t2 = __shfl(xf2, n, 32);        // xf2 of leaves 0..15
            b0 = lo ? xf0 : t2;                   // k=0 (lo) / k=2 (hi)
            b1 = lo ? xf1 : 1.0f;                 // k=1 (lo) / k=3 (hi)
        } else {
            float t0 = __shfl(xf0, 16 + n, 32);   // xf0 of leaves 16..31
            float t1 = __shfl(xf1, 16 + n, 32);   // xf1 of leaves 16..31
            b0 = lo ? t0 : xf2;
            b1 = lo ? t1 : 1.0f;
        }
        v2f B; B.x = b0; B.y = b1;

        v8f C = {};
        // 8 args: (neg_a, A, neg_b, B, c_mod, C, reuse_a, reuse_b)
        v8f D = __builtin_amdgcn_wmma_f32_16x16x4_f32(
            false, A, false, B, (short)0, C, false, false);

        // D row r lives in VGPR r across lanes 0-15; rows 0..2 are x,y,z of
        // leaf n -> 12 contiguous bytes per lane (merge to global_store_b96).
        if (lo) {
            unsigned leaf = waveBase + (unsigned)(g * 16 + n);
            float* o = out + 3u * leaf;
            o[0] = D[0];
            o[1] = D[1];
            o[2] = D[2];
        }
    }
}

extern "C" void kernel_launch(void* const* d_in, const int* in_sizes, int n_in,
                              void* d_out, int out_size, void* d_ws, size_t ws_size,
                              hipStream_t stream) {
    const int*   flags  = (const int*)d_in[0];    // int32 [349525]
    const float* offset = (const float*)d_in[1];  // f32 [3]
    const float* scale  = (const float*)d_in[2];  // f32 [3]
    float*       out    = (float*)d_out;          // f32 [1048576*3]

    (void)in_sizes; (void)n_in; (void)out_size; (void)d_ws; (void)ws_size;

    dim3 grid(N_POINTS / BLOCK), block(BLOCK);
    nbit_tree_decode_wmma<<<grid, block, 0, stream>>>(flags, offset, scale, out);
}